// AttentionModule_3341484556477
// MI455X (gfx1250) — compile-verified
//
#include <hip/hip_runtime.h>
#include <math.h>

// Problem constants (from reference setup_inputs)
#define S_LEN 1024
#define D_DIM 4096
#define NH    32
#define HDIM  128
#define CTX   2048

typedef float v2f __attribute__((ext_vector_type(2)));
typedef float v8f __attribute__((ext_vector_type(8)));

// ---------------------------------------------------------------------------
// RMSNorm: one block per row of xs
// ---------------------------------------------------------------------------
__global__ __launch_bounds__(256) void rmsnorm_kernel(const float* __restrict__ x,
                                                      const float* __restrict__ w,
                                                      float* __restrict__ xn) {
  __shared__ float red[256];
  const int row = blockIdx.x;
  const float* xr = x + (size_t)row * D_DIM;
  float s = 0.f;
  for (int i = threadIdx.x; i < D_DIM; i += 256) { float v = xr[i]; s += v * v; }
  red[threadIdx.x] = s;
  __syncthreads();
  for (int off = 128; off > 0; off >>= 1) {
    if (threadIdx.x < off) red[threadIdx.x] += red[threadIdx.x + off];
    __syncthreads();
  }
  const float scale = rsqrtf(red[0] * (1.0f / D_DIM) + 1e-6f);
  for (int i = threadIdx.x; i < D_DIM; i += 256)
    xn[(size_t)row * D_DIM + i] = xr[i] * scale * w[i];
}

// ---------------------------------------------------------------------------
// Out[M,N] = A[M,K] @ W[N,K]^T  via V_WMMA_F32_16X16X4_F32
// Block: 256 threads (8 waves). Block tile 64(M) x 128(N), K chunk 32.
// Double-buffered LDS staging, one barrier per K chunk.
// f32 WMMA operand layout (ISA 7.12.2):
//   A: lane L (L<16) holds row L, K = {0,1}; lane L+16 holds row L, K = {2,3}
//   B: lane n (n<16) holds col n, K = {0,1}; lane n+16 holds col n, K = {2,3}
//   C/D: lane n, VGPR j -> (M = j + 8*(lane>=16), N = n)
// ---------------------------------------------------------------------------
#define BM 64
#define BN 128
#define BK 32
#define LPAD 36  // padded K-chunk row stride (multiple of 4 -> 16B aligned, conflict-free)

__global__ __launch_bounds__(256) void gemm_awt_kernel(const float* __restrict__ A,
                                                       const float* __restrict__ W,
                                                       float* __restrict__ Out,
                                                       int M, int N, int K) {
  __shared__ float As[2][BM][LPAD];
  __shared__ float Ws[2][BN][LPAD];

  const int tid = threadIdx.x;
  const int wave = tid >> 5;
  const int lane = tid & 31;
  const int g = lane >> 4;   // K pair group: 0 -> K{0,1}, 1 -> K{2,3}
  const int ln = lane & 15;

  const int bm = blockIdx.x * BM;
  const int bn = blockIdx.y * BN;
  const int wm = (wave >> 2) * 32;  // 0 or 32
  const int wn = (wave & 3) * 32;   // 0..96

  const int sr = tid >> 3;         // staging row base (0..31), 8 float4 per BK row
  const int scol = (tid & 7) * 4;  // staging col (0..28)

  v8f acc[2][2] = {};
  float4 areg[2], wreg[4];

  // Prologue: fetch + stage chunk 0 into buffer 0
#pragma unroll
  for (int i = 0; i < 2; ++i)
    areg[i] = *(const float4*)(A + (size_t)(bm + sr + 32 * i) * K + scol);
#pragma unroll
  for (int i = 0; i < 4; ++i)
    wreg[i] = *(const float4*)(W + (size_t)(bn + sr + 32 * i) * K + scol);
#pragma unroll
  for (int i = 0; i < 2; ++i) *(float4*)&As[0][sr + 32 * i][scol] = areg[i];
#pragma unroll
  for (int i = 0; i < 4; ++i) *(float4*)&Ws[0][sr + 32 * i][scol] = wreg[i];
  __syncthreads();

  const int nch = K / BK;
  for (int ch = 0; ch < nch; ++ch) {
    const int cur = ch & 1;
    // Fetch next chunk into registers (overlaps with WMMA compute below)
    if (ch + 1 < nch) {
      const int k0 = (ch + 1) * BK;
#pragma unroll
      for (int i = 0; i < 2; ++i)
        areg[i] = *(const float4*)(A + (size_t)(bm + sr + 32 * i) * K + k0 + scol);
#pragma unroll
      for (int i = 0; i < 4; ++i)
        wreg[i] = *(const float4*)(W + (size_t)(bn + sr + 32 * i) * K + k0 + scol);
    }

#pragma unroll
    for (int kk = 0; kk < BK; kk += 4) {
      const int kc = kk + 2 * g;
      v2f a0 = *(const v2f*)&As[cur][wm + ln][kc];
      v2f a1 = *(const v2f*)&As[cur][wm + 16 + ln][kc];
      v2f b0 = *(const v2f*)&Ws[cur][wn + ln][kc];
      v2f b1 = *(const v2f*)&Ws[cur][wn + 16 + ln][kc];
      acc[0][0] = __builtin_amdgcn_wmma_f32_16x16x4_f32(false, a0, false, b0, (short)0,
                                                        acc[0][0], false, false);
      acc[0][1] = __builtin_amdgcn_wmma_f32_16x16x4_f32(false, a0, false, b1, (short)0,
                                                        acc[0][1], false, false);
      acc[1][0] = __builtin_amdgcn_wmma_f32_16x16x4_f32(false, a1, false, b0, (short)0,
                                                        acc[1][0], false, false);
      acc[1][1] = __builtin_amdgcn_wmma_f32_16x16x4_f32(false, a1, false, b1, (short)0,
                                                        acc[1][1], false, false);
    }

    if (ch + 1 < nch) {
#pragma unroll
      for (int i = 0; i < 2; ++i) *(float4*)&As[cur ^ 1][sr + 32 * i][scol] = areg[i];
#pragma unroll
      for (int i = 0; i < 4; ++i) *(float4*)&Ws[cur ^ 1][sr + 32 * i][scol] = wreg[i];
      __syncthreads();
    }
  }

#pragma unroll
  for (int mi = 0; mi < 2; ++mi)
#pragma unroll
    for (int ni = 0; ni < 2; ++ni)
#pragma unroll
      for (int j = 0; j < 8; ++j) {
        const int row = bm + wm + mi * 16 + j + 8 * g;
        const int col = bn + wn + ni * 16 + ln;
        Out[(size_t)row * N + col] = acc[mi][ni][j];
      }
}

// ---------------------------------------------------------------------------
// RoPE in place on a (S, H, HD) fp32 tensor, position offset = CTX
// ---------------------------------------------------------------------------
__global__ __launch_bounds__(256) void rope_kernel(float* __restrict__ x, int offset) {
  const int idx = blockIdx.x * 256 + threadIdx.x;  // S*H*(HD/2)
  if (idx >= S_LEN * NH * (HDIM / 2)) return;
  const int d = idx & 63;
  const int sh = idx >> 6;       // s*NH + h
  const int s = sh >> 5;         // / NH
  // inv_freq = 10000^(-d/64) = exp2(-d * log2(10000)/64)
  const float inv_freq = exp2f(-(float)d * (13.287712379549449f / 64.0f));
  const float ang = (float)(s + offset) * inv_freq;
  float sn, cs;
  sincosf(ang, &sn, &cs);
  float* p = x + (size_t)sh * HDIM + d;
  const float x1 = p[0], x2 = p[64];
  p[0] = x1 * cs - x2 * sn;
  p[64] = x2 * cs + x1 * sn;
}

// ---------------------------------------------------------------------------
// Flash attention. Block = 8 waves, head = blockIdx.y, wave handles the
// 16-query tile  mtile = blockIdx.x*8 + wave.  Keys processed 16 at a time;
// K/V tiles double-buffered in LDS (one barrier per key tile).
// Online softmax in registers.
// ---------------------------------------------------------------------------
__global__ __launch_bounds__(256) void attn_kernel(const float* __restrict__ q,
                                                   const float* __restrict__ kn,
                                                   const float* __restrict__ vn,
                                                   const float* __restrict__ ck,
                                                   const float* __restrict__ cv,
                                                   float* __restrict__ o) {
  __shared__ float Ks[2][16][132];  // 16 keys x 128 dims, padded (conflict-free)
  __shared__ float Vs[2][16][132];
  __shared__ float Ps[8][16][18];   // per-wave P transpose buffer (even stride -> b64 ok)

  const int tid = threadIdx.x;
  const int wave = tid >> 5;
  const int lane = tid & 31;
  const int g = lane >> 4;
  const int ln = lane & 15;
  const int h = blockIdx.y;
  const int mbase = (blockIdx.x * 8 + wave) * 16;

  // Staging map: 16 rows x 128 floats = 512 float4; 2 per thread.
  // Thread loads rows (wave + 8*i), cols lane*4 .. lane*4+3  (coalesced).
  const int svcol = lane * 4;

  // Preload q tile (16 queries x 128 dims) in WMMA-A layout: 32 K-steps of v2f
  v2f qreg[32];
  const float* qp = q + (size_t)(mbase + ln) * D_DIM + h * HDIM + 2 * g;
#pragma unroll
  for (int i = 0; i < 32; ++i) qreg[i] = *(const v2f*)(qp + 4 * i);

  v8f acc[8] = {};
  float mrow[8], lsum[8];
#pragma unroll
  for (int j = 0; j < 8; ++j) { mrow[j] = -1e30f; lsum[j] = 0.f; }

  // All waves in the block run the same trip count (for the highest mtile);
  // trailing tiles of lower mtiles are fully masked and contribute nothing.
  const int ntiles = CTX / 16 + blockIdx.x * 8 + 8;
  const float scale = 0.08838834764831845f;  // 1/sqrt(128)

  float4 kreg[2], vreg[2];

  // Prologue: fetch + stage key tile 0 into buffer 0
#pragma unroll
  for (int i = 0; i < 2; ++i) {
    const int t = wave + 8 * i;  // tbase = 0 -> always cache side (t < CTX)
    kreg[i] = *(const float4*)(ck + (size_t)t * D_DIM + h * HDIM + svcol);
    vreg[i] = *(const float4*)(cv + (size_t)t * D_DIM + h * HDIM + svcol);
  }
#pragma unroll
  for (int i = 0; i < 2; ++i) {
    *(float4*)&Ks[0][wave + 8 * i][svcol] = kreg[i];
    *(float4*)&Vs[0][wave + 8 * i][svcol] = vreg[i];
  }
  __syncthreads();

  for (int nt = 0; nt < ntiles; ++nt) {
    const int cur = nt & 1;
    const int tbase = nt * 16;

    // Fetch next key tile into registers (overlaps with compute)
    if (nt + 1 < ntiles) {
      const int tb1 = tbase + 16;
#pragma unroll
      for (int i = 0; i < 2; ++i) {
        const int t = tb1 + wave + 8 * i;
        const float* ksrc = (t < CTX) ? (ck + (size_t)t * D_DIM + h * HDIM + svcol)
                                      : (kn + (size_t)(t - CTX) * D_DIM + h * HDIM + svcol);
        const float* vsrc = (t < CTX) ? (cv + (size_t)t * D_DIM + h * HDIM + svcol)
                                      : (vn + (size_t)(t - CTX) * D_DIM + h * HDIM + svcol);
        kreg[i] = *(const float4*)ksrc;
        vreg[i] = *(const float4*)vsrc;
      }
    }

    // Scores tile: 16 queries x 16 keys, reduce over 128 head dims
    v8f sc = {};
#pragma unroll
    for (int i = 0; i < 32; ++i) {
      v2f b = *(const v2f*)&Ks[cur][ln][4 * i + 2 * g];
      sc = __builtin_amdgcn_wmma_f32_16x16x4_f32(false, qreg[i], false, b, (short)0, sc,
                                                 false, false);
    }

    // Scale + causal mask + online softmax (row = query, spread over VGPRs;
    // the 16 lanes of each half-wave hold the 16 key columns of a row)
    const int tcol = tbase + ln;
    float fs[8];
#pragma unroll
    for (int j = 0; j < 8; ++j) {
      const int grow = mbase + j + 8 * g;
      float sv = (tcol <= grow + CTX) ? sc[j] * scale : -1e30f;
      float rm = sv;
      rm = fmaxf(rm, __shfl_xor(rm, 1, 32));
      rm = fmaxf(rm, __shfl_xor(rm, 2, 32));
      rm = fmaxf(rm, __shfl_xor(rm, 4, 32));
      rm = fmaxf(rm, __shfl_xor(rm, 8, 32));
      const float mnew = fmaxf(mrow[j], rm);
      const float f = __expf(mrow[j] - mnew);
      mrow[j] = mnew;
      const float p = __expf(sv - mnew);
      float ps = p;
      ps += __shfl_xor(ps, 1, 32);
      ps += __shfl_xor(ps, 2, 32);
      ps += __shfl_xor(ps, 4, 32);
      ps += __shfl_xor(ps, 8, 32);
      lsum[j] = lsum[j] * f + ps;
      fs[j] = f;
      Ps[wave][j + 8 * g][ln] = p;   // D-layout -> row-major transpose via LDS
    }
#pragma unroll
    for (int t8 = 0; t8 < 8; ++t8)
#pragma unroll
      for (int j = 0; j < 8; ++j) acc[t8][j] *= fs[j];

    // Same-wave LDS RAW; DS ops are in-order per wave, fence for safety
    asm volatile("s_wait_dscnt 0" ::: "memory");

    // acc(16x128) += P(16x16) @ V(16x128)
#pragma unroll
    for (int kk = 0; kk < 16; kk += 4) {
      v2f a = *(const v2f*)&Ps[wave][ln][kk + 2 * g];
#pragma unroll
      for (int nti = 0; nti < 8; ++nti) {
        v2f b;
        b.x = Vs[cur][kk + 2 * g][nti * 16 + ln];
        b.y = Vs[cur][kk + 2 * g + 1][nti * 16 + ln];
        acc[nti] = __builtin_amdgcn_wmma_f32_16x16x4_f32(false, a, false, b, (short)0,
                                                         acc[nti], false, false);
      }
    }

    // Stage next tile into the other buffer; single barrier per iteration.
    if (nt + 1 < ntiles) {
#pragma unroll
      for (int i = 0; i < 2; ++i) {
        *(float4*)&Ks[cur ^ 1][wave + 8 * i][svcol] = kreg[i];
        *(float4*)&Vs[cur ^ 1][wave + 8 * i][svcol] = vreg[i];
      }
      __syncthreads();
    }
  }

  // Normalize and write: o[s][h*128 + d]
#pragma unroll
  for (int nti = 0; nti < 8; ++nti)
#pragma unroll
    for (int j = 0; j < 8; ++j) {
      const int row = mbase + j + 8 * g;
      const int col = h * HDIM + nti * 16 + ln;
      o[(size_t)row * D_DIM + col] = acc[nti][j] / lsum[j];
    }
}

// ---------------------------------------------------------------------------
extern "C" void kernel_launch(void* const* d_in, const int* in_sizes, int n_in,
                              void* d_out, int out_size, void* d_ws, size_t ws_size,
                              hipStream_t stream) {
  (void)in_sizes; (void)n_in; (void)out_size; (void)ws_size;
  const float* xs      = (const float*)d_in[0];
  const float* cache_k = (const float*)d_in[1];
  const float* cache_v = (const float*)d_in[2];
  const float* norm_w  = (const float*)d_in[3];
  const float* wq      = (const float*)d_in[4];
  const float* wk      = (const float*)d_in[5];
  const float* wv      = (const float*)d_in[6];
  const float* wo      = (const float*)d_in[7];
  float* out = (float*)d_out;

  const size_t SD = (size_t)S_LEN * D_DIM;
  float* ws = (float*)d_ws;
  float* xn = ws;            // S*D
  float* qb = ws + SD;       // S*D  (q after projection/rope)
  float* kb = ws + 2 * SD;   // S*D  (k_new)
  float* vb = ws + 3 * SD;   // S*D  (v_new)
  float* at = xn;            // attention output reuses xn buffer

  rmsnorm_kernel<<<S_LEN, 256, 0, stream>>>(xs, norm_w, xn);

  dim3 gg(S_LEN / BM, D_DIM / BN);
  gemm_awt_kernel<<<gg, 256, 0, stream>>>(xn, wq, qb, S_LEN, D_DIM, D_DIM);
  gemm_awt_kernel<<<gg, 256, 0, stream>>>(xn, wk, kb, S_LEN, D_DIM, D_DIM);
  gemm_awt_kernel<<<gg, 256, 0, stream>>>(xn, wv, vb, S_LEN, D_DIM, D_DIM);

  const int nrope = S_LEN * NH * (HDIM / 2);
  rope_kernel<<<(nrope + 255) / 256, 256, 0, stream>>>(qb, CTX);
  rope_kernel<<<(nrope + 255) / 256, 256, 0, stream>>>(kb, CTX);

  dim3 ga(S_LEN / 16 / 8, NH);
  attn_kernel<<<ga, 256, 0, stream>>>(qb, kb, vb, cache_k, cache_v, at);

  gemm_awt_kernel<<<gg, 256, 0, stream>>>(at, wo, out, S_LEN, D_DIM, D_DIM);
}